// ConnCompAttention_83760452206644
// MI455X (gfx1250) — compile-verified
//
#include <hip/hip_runtime.h>

#define H_IMG 160
#define W_IMG 160
#define CIN   256
#define BATCH 8
#define NPIX  (H_IMG * W_IMG)          // 25600
#define LIMIT 20
#define LISTCAP 32768
#define INTMAXV 0x7FFFFFFF
#define CHUNK 16                        // input channels per LDS chunk

typedef float v2f __attribute__((ext_vector_type(2)));
typedef float v8f __attribute__((ext_vector_type(8)));

#ifndef __has_builtin
#define __has_builtin(x) 0
#endif

#if __has_builtin(__builtin_amdgcn_global_load_async_to_lds_b32)
#define USE_ASYNC_LDS 1
#else
#define USE_ASYNC_LDS 0
#endif

// Builtin signature (from hipcc diagnostic): param0 = AS1 ("__device__") int*,
// param1 = AS3 ("__shared__") int*, then imm offset, imm cpol.
#define AS1_INT(p) ((__attribute__((address_space(1))) int*)(p))
#define AS3_INT(p) ((__attribute__((address_space(3))) int*)(p))

__device__ __forceinline__ void async_wait0()
{
#if USE_ASYNC_LDS
#if __has_builtin(__builtin_amdgcn_s_wait_asynccnt)
    __builtin_amdgcn_s_wait_asynccnt(0);
#else
    asm volatile("s_wait_asynccnt 0" ::: "memory");
#endif
#endif
}

// ---------------------------------------------------------------------------
// Kernel 1: 3x3 conv (256 -> 3) as implicit GEMM on V_WMMA_F32_16X16X4_F32,
// then bias + softmax over 3 channels + argmax.
// Block = 160 threads (5 waves), tile = 80 pixels of one output row.
// Double-buffered LDS chunks; staging via GLOBAL_LOAD_ASYNC_TO_LDS_B32
// (ASYNCcnt) overlapped with the WMMA loop.
// ---------------------------------------------------------------------------
__global__ __launch_bounds__(160)
void conv_attn_kernel(const float* __restrict__ x, const float* __restrict__ w,
                      const float* __restrict__ bias,
                      float* __restrict__ out_attn, unsigned char* __restrict__ am)
{
    const int tile = blockIdx.x;           // 0..1 -> 80 px each
    const int y    = blockIdx.y;           // output row
    const int b    = blockIdx.z;
    const int tid  = threadIdx.x;          // 0..159
    const int wv   = tid >> 5;             // wave 0..4
    const int lane = tid & 31;
    const int laneM = lane & 15;
    const int hi    = lane >> 4;           // 0: K0/K1  1: K2/K3
    const int xbase = tile * 80;

    __shared__ float xs[2][3 * 82][17];    // [buf][dy*82+col][channel]
    __shared__ float wl[2][27][17];        // [buf][o*9+tap][channel]
    __shared__ float stage[5][16][17];     // D re-layout per wave

    // stage channel chunk `ch` (CHUNK channels) into buffer `buf`
    auto stage_chunk = [&](int ch, int buf) {
        const int cbase = ch * CHUNK;
        for (int i = tid; i < CHUNK * 3 * 82; i += 160) {
            int c   = i / (3 * 82);
            int rem = i - c * (3 * 82);
            int row = rem / 82;
            int col = rem - row * 82;
            int gy = y + row - 1;
            int gx = xbase + col - 1;
            bool in = ((unsigned)gy < H_IMG) && ((unsigned)gx < W_IMG);
#if USE_ASYNC_LDS
            if (in) {
                const float* src =
                    &x[(((size_t)b * CIN + (cbase + c)) * H_IMG + gy) * W_IMG + gx];
                __builtin_amdgcn_global_load_async_to_lds_b32(
                    AS1_INT(src), AS3_INT(&xs[buf][row * 82 + col][c]), 0, 0);
            } else {
                xs[buf][row * 82 + col][c] = 0.f;   // halo zero: disjoint from async dst
            }
#else
            float v = 0.f;
            if (in)
                v = x[(((size_t)b * CIN + (cbase + c)) * H_IMG + gy) * W_IMG + gx];
            xs[buf][row * 82 + col][c] = v;
#endif
        }
        for (int i = tid; i < 27 * CHUNK; i += 160) {
            int c  = i & (CHUNK - 1);
            int ot = i >> 4;               // o*9 + tap
            int o = ot / 9, tap = ot - o * 9;
            const float* src = &w[((size_t)o * CIN + (cbase + c)) * 9 + tap];
#if USE_ASYNC_LDS
            __builtin_amdgcn_global_load_async_to_lds_b32(
                AS1_INT(src), AS3_INT(&wl[buf][ot][c]), 0, 0);
#else
            wl[buf][ot][c] = *src;
#endif
        }
    };

    v8f acc = {};                          // 16x16 fp32 accumulator

    stage_chunk(0, 0);
    async_wait0();
    __syncthreads();

    for (int ch = 0; ch < CIN / CHUNK; ++ch) {   // 16 chunks
        const int cur = ch & 1;
        if (ch + 1 < CIN / CHUNK)
            stage_chunk(ch + 1, cur ^ 1);        // overlap next-chunk DMA with WMMA

        for (int tap = 0; tap < 9; ++tap) {
            int dy = tap / 3, dx = tap - dy * 3;
            const float* arow = &xs[cur][dy * 82 + (wv * 16 + laneM + dx)][0];
            const float* brow = (laneM < 3) ? &wl[cur][laneM * 9 + tap][0]
                                            : (const float*)0;
            #pragma unroll
            for (int c0 = 0; c0 < CHUNK; c0 += 4) {
                int ca = c0 + 2 * hi;
                v2f a;  a.x = arow[ca];  a.y = arow[ca + 1];
                v2f bb;
                if (brow) { bb.x = brow[ca]; bb.y = brow[ca + 1]; }
                else      { bb.x = 0.f;     bb.y = 0.f; }
                acc = __builtin_amdgcn_wmma_f32_16x16x4_f32(
                          false, a, false, bb, (short)0, acc, false, false);
            }
        }
        async_wait0();       // next-chunk LDS writes landed
        __syncthreads();     // all waves done reading cur / writing cur^1
    }

    // D layout: vgpr r -> M = r + 8*hi, N = laneM.  Regroup via LDS.
    #pragma unroll
    for (int r = 0; r < 8; ++r)
        stage[wv][r + 8 * hi][laneM] = acc[r];
    __syncthreads();

    if (hi == 0) {
        int px = xbase + wv * 16 + laneM;
        float l0 = stage[wv][laneM][0] + bias[0];
        float l1 = stage[wv][laneM][1] + bias[1];
        float l2 = stage[wv][laneM][2] + bias[2];
        float mx = fmaxf(l0, fmaxf(l1, l2));
        float e0 = __expf(l0 - mx), e1 = __expf(l1 - mx), e2 = __expf(l2 - mx);
        float inv = 1.f / (e0 + e1 + e2);
        size_t base = (size_t)b * 3 * NPIX + (size_t)y * W_IMG + px;
        out_attn[base]            = e0 * inv;
        out_attn[base + NPIX]     = e1 * inv;
        out_attn[base + 2 * NPIX] = e2 * inv;
        int amv = 0; float bv = l0;
        if (l1 > bv) { amv = 1; bv = l1; }
        if (l2 > bv) { amv = 2; }
        am[(size_t)b * NPIX + y * W_IMG + px] = (unsigned char)amv;
    }
}

// ---------------------------------------------------------------------------
// CCL: labels[(m*BATCH+b)*NPIX + p] ; label = 1 + max flat index in component.
// ---------------------------------------------------------------------------
__global__ void ccl_init_kernel(const unsigned char* __restrict__ am,
                                int* __restrict__ labels)
{
    int idx = blockIdx.x * 256 + threadIdx.x;
    if (idx >= 2 * BATCH * NPIX) return;
    int mb = idx / NPIX, p = idx - mb * NPIX;
    int m = mb / BATCH, b = mb - m * BATCH;
    labels[idx] = (am[(size_t)b * NPIX + p] == (unsigned char)(m + 1)) ? (p + 1) : 0;
}

__global__ __launch_bounds__(256)
void ccl_prop_kernel(int* __restrict__ labels)
{
    int mb = blockIdx.z;
    int x0 = blockIdx.x * 32, y0 = blockIdx.y * 32;
    int tx = threadIdx.x, ty = threadIdx.y;         // (32, 8)
    int tid = ty * 32 + tx;
    __shared__ int lab[34][34];
    int* L = labels + (size_t)mb * NPIX;
    for (int i = tid; i < 34 * 34; i += 256) {
        int ly = i / 34, lx = i - ly * 34;
        int gy = y0 + ly - 1, gx = x0 + lx - 1;
        int v = 0;
        if ((unsigned)gy < H_IMG && (unsigned)gx < W_IMG) v = L[gy * W_IMG + gx];
        lab[ly][lx] = v;
    }
    __syncthreads();
    for (int it = 0; it < 48; ++it) {
        #pragma unroll
        for (int k = 0; k < 4; ++k) {
            int r = ty + 8 * k, c = tx;
            int v = lab[r + 1][c + 1];
            if (v > 0) {
                v = max(v, lab[r][c + 1]); v = max(v, lab[r + 2][c + 1]);
                v = max(v, lab[r + 1][c]); v = max(v, lab[r + 1][c + 2]);
                lab[r + 1][c + 1] = v;     // monotone in-place max: race-safe
            }
        }
        __syncthreads();
    }
    for (int k = 0; k < 4; ++k) {
        int r = ty + 8 * k;
        L[(y0 + r) * W_IMG + (x0 + tx)] = lab[r + 1][tx + 1];
    }
}

// ---------------------------------------------------------------------------
// Representatives: pixel p with label == p+1 is its component's unique rep.
// ---------------------------------------------------------------------------
__global__ void reps_kernel(const int* __restrict__ labels, int* __restrict__ lists,
                            int* __restrict__ listcnt, int* __restrict__ bgflag)
{
    int idx = blockIdx.x * 256 + threadIdx.x;
    if (idx >= 2 * BATCH * NPIX) return;
    int mb = idx / NPIX, p = idx - mb * NPIX;
    int l = labels[idx];
    if (l == 0) {
        bgflag[mb] = 1;                                // idempotent store
    } else if (l == p + 1) {
        int k = atomicAdd(&listcnt[mb], 1);
        if (k < LISTCAP) lists[(size_t)mb * LISTCAP + k] = l;
    }
}

// ---------------------------------------------------------------------------
// 20 smallest unique labels (ascending), rank 0 = background label 0 if present.
// ---------------------------------------------------------------------------
__global__ __launch_bounds__(256)
void topk_kernel(const int* __restrict__ lists, const int* __restrict__ listcnt,
                 const int* __restrict__ bgflag, int* __restrict__ slots)
{
    int mb = blockIdx.x;
    const int* lst = lists + (size_t)mb * LISTCAP;
    int n = min(listcnt[mb], LISTCAP);
    __shared__ int red[256];
    int tid = threadIdx.x;
    int kstart = bgflag[mb] ? 1 : 0;
    if (tid == 0 && kstart == 1) slots[mb * LIMIT] = 0;
    int prev = 0;
    for (int k = kstart; k < LIMIT; ++k) {
        int local = INTMAXV;
        for (int i = tid; i < n; i += 256) {
            int v = lst[i];
            if (v > prev && v < local) local = v;
        }
        red[tid] = local;
        __syncthreads();
        for (int s = 128; s > 0; s >>= 1) {
            if (tid < s) red[tid] = min(red[tid], red[tid + s]);
            __syncthreads();
        }
        int m = red[0];
        __syncthreads();
        if (tid == 0) slots[mb * LIMIT + k] = m;
        prev = m;
        if (m == INTMAXV) {
            for (int kk = k + 1 + tid; kk < LIMIT; kk += 256)
                slots[mb * LIMIT + kk] = INTMAXV;
            break;
        }
    }
}

__global__ void seg_kernel(const int* __restrict__ labels, const int* __restrict__ slots,
                           unsigned char* __restrict__ seg)
{
    int idx = blockIdx.x * 256 + threadIdx.x;
    if (idx >= 2 * BATCH * NPIX) return;
    int mb = idx / NPIX;
    int l = labels[idx];
    unsigned char s = LIMIT;
    if (l > 0) {
        const int* sl = slots + mb * LIMIT;
        for (int k = 0; k < LIMIT; ++k)
            if (sl[k] == l) { s = (unsigned char)k; break; }
    }
    seg[idx] = s;
}

// ---------------------------------------------------------------------------
// Segmented sums: per (mask,b,rank) channel sums + counts via LDS atomics.
// Block = 256 threads, 256 pixels; channels in 4 chunks of 64.
// ---------------------------------------------------------------------------
__global__ __launch_bounds__(256)
void stats_kernel(const float* __restrict__ x, const unsigned char* __restrict__ seg,
                  float* __restrict__ sums, float* __restrict__ cnts)
{
    int b = blockIdx.y;
    int p = blockIdx.x * 256 + threadIdx.x;
    int tid = threadIdx.x;
    int s1 = seg[(size_t)b * NPIX + p];
    int s2 = seg[((size_t)BATCH + b) * NPIX + p];
    __shared__ float acc[2][LIMIT + 1][64];
    for (int cc = 0; cc < 4; ++cc) {
        for (int i = tid; i < 2 * (LIMIT + 1) * 64; i += 256) ((float*)acc)[i] = 0.f;
        __syncthreads();
        int cbase = cc * 64;
        for (int c = 0; c < 64; ++c) {
            float v = x[((size_t)b * CIN + cbase + c) * NPIX + p];
            atomicAdd(&acc[0][s1][c], v);
            atomicAdd(&acc[1][s2][c], v);
        }
        __syncthreads();
        for (int i = tid; i < 2 * LIMIT * 64; i += 256) {
            int m = i / (LIMIT * 64);
            int rem = i - m * (LIMIT * 64);
            int sg = rem / 64, c = rem - sg * 64;
            float v = acc[m][sg][c];
            if (v != 0.f)
                atomicAdd(&sums[(((size_t)m * BATCH + b) * LIMIT + sg) * CIN + cbase + c], v);
        }
        __syncthreads();
    }
    __shared__ float hcnt[2][LIMIT + 1];
    for (int i = tid; i < 2 * (LIMIT + 1); i += 256) ((float*)hcnt)[i] = 0.f;
    __syncthreads();
    atomicAdd(&hcnt[0][s1], 1.f);
    atomicAdd(&hcnt[1][s2], 1.f);
    __syncthreads();
    for (int i = tid; i < 2 * LIMIT; i += 256) {
        int m = i / LIMIT, sg = i - m * LIMIT;
        float v = hcnt[m][sg];
        if (v != 0.f) atomicAdd(&cnts[((size_t)m * BATCH + b) * LIMIT + sg], v);
    }
}

// ---------------------------------------------------------------------------
// means, cross-Gram M = m1 @ m2^T, pf = 1 + M (valid pairs), row/col products.
// ---------------------------------------------------------------------------
__global__ __launch_bounds__(256)
void meansmult_kernel(const float* __restrict__ sums, const float* __restrict__ cnts,
                      float* __restrict__ multg)
{
    int b = blockIdx.x;
    int tid = threadIdx.x;
    __shared__ float m1[LIMIT][CIN], m2[LIMIT][CIN];
    __shared__ float pf[LIMIT][LIMIT];
    __shared__ float c1[LIMIT], c2[LIMIT];
    for (int i = tid; i < LIMIT; i += 256) {
        c1[i] = cnts[((size_t)0 * BATCH + b) * LIMIT + i];
        c2[i] = cnts[((size_t)1 * BATCH + b) * LIMIT + i];
    }
    __syncthreads();
    for (int i = tid; i < LIMIT * CIN; i += 256) {
        int sg = i >> 8, c = i & 255;
        m1[sg][c] = sums[(((size_t)0 * BATCH + b) * LIMIT + sg) * CIN + c] / fmaxf(c1[sg], 1.f);
        m2[sg][c] = sums[(((size_t)1 * BATCH + b) * LIMIT + sg) * CIN + c] / fmaxf(c2[sg], 1.f);
    }
    __syncthreads();
    for (int idx = tid; idx < LIMIT * LIMIT; idx += 256) {
        int i = idx / LIMIT, j = idx - i * LIMIT;
        float d = 0.f;
        for (int c = 0; c < CIN; ++c) d += m1[i][c] * m2[j][c];
        pf[i][j] = (c1[i] > 0.f && c2[j] > 0.f) ? (1.f + d) : 1.f;
    }
    __syncthreads();
    if (tid < LIMIT) {
        float p1 = 1.f, p2 = 1.f;
        for (int j = 0; j < LIMIT; ++j) { p1 *= pf[tid][j]; p2 *= pf[j][tid]; }
        multg[((size_t)0 * BATCH + b) * (LIMIT + 1) + tid] = p1;
        multg[((size_t)1 * BATCH + b) * (LIMIT + 1) + tid] = p2;
    }
    if (tid == 0) {
        multg[((size_t)0 * BATCH + b) * (LIMIT + 1) + LIMIT] = 1.f;
        multg[((size_t)1 * BATCH + b) * (LIMIT + 1) + LIMIT] = 1.f;
    }
}

__global__ void factor_kernel(const unsigned char* __restrict__ seg,
                              const float* __restrict__ multg, float* __restrict__ factor)
{
    int idx = blockIdx.x * 256 + threadIdx.x;
    if (idx >= BATCH * NPIX) return;
    int b = idx / NPIX;
    int s1 = seg[idx];
    int s2 = seg[(size_t)BATCH * NPIX + idx];
    factor[idx] = multg[((size_t)0 * BATCH + b) * (LIMIT + 1) + s1] *
                  multg[((size_t)1 * BATCH + b) * (LIMIT + 1) + s2];
}

// ---------------------------------------------------------------------------
// BatchNorm batch stats over x2 = x * factor (x2 never materialized).
// ---------------------------------------------------------------------------
__global__ __launch_bounds__(256)
void chanstat_kernel(const float* __restrict__ x, const float* __restrict__ factor,
                     float* __restrict__ chansum, float* __restrict__ chansq)
{
    int c = blockIdx.x, b = blockIdx.y;
    const float* xp = x + ((size_t)b * CIN + c) * NPIX;
    const float* fp = factor + (size_t)b * NPIX;
    float s = 0.f, q = 0.f;
    for (int p = threadIdx.x; p < NPIX; p += 256) {
        float v = xp[p] * fp[p];
        s += v; q += v * v;
    }
    __shared__ float rs[256], rq[256];
    rs[threadIdx.x] = s; rq[threadIdx.x] = q;
    __syncthreads();
    for (int st = 128; st > 0; st >>= 1) {
        if (threadIdx.x < st) {
            rs[threadIdx.x] += rs[threadIdx.x + st];
            rq[threadIdx.x] += rq[threadIdx.x + st];
        }
        __syncthreads();
    }
    if (threadIdx.x == 0) { atomicAdd(&chansum[c], rs[0]); atomicAdd(&chansq[c], rq[0]); }
}

__global__ void finalize_kernel(const float* __restrict__ chansum,
                                const float* __restrict__ chansq, float* __restrict__ meaninv)
{
    int c = threadIdx.x;
    float N = (float)((size_t)BATCH * NPIX);
    float mean = chansum[c] / N;
    float var = chansq[c] / N - mean * mean;
    meaninv[c] = mean;
    meaninv[CIN + c] = rsqrtf(var + 1e-5f);
}

__global__ __launch_bounds__(256)
void norm_kernel(const float* __restrict__ x, const float* __restrict__ factor,
                 const float* __restrict__ meaninv, const float* __restrict__ gamma,
                 const float* __restrict__ beta, float* __restrict__ out)
{
    int c = blockIdx.x, b = blockIdx.y;
    float mean = meaninv[c], inv = meaninv[CIN + c];
    float g = gamma[c], bt = beta[c];
    const float* xp = x + ((size_t)b * CIN + c) * NPIX;
    const float* fp = factor + (size_t)b * NPIX;
    float* op = out + ((size_t)b * CIN + c) * NPIX;
    for (int p = threadIdx.x; p < NPIX; p += 256) {
        __builtin_prefetch(xp + p + 2048, 0, 0);   // global_prefetch_b8
        op[p] = (xp[p] * fp[p] - mean) * inv * g + bt;
    }
}

// ---------------------------------------------------------------------------
extern "C" void kernel_launch(void* const* d_in, const int* in_sizes, int n_in,
                              void* d_out, int out_size, void* d_ws, size_t ws_size,
                              hipStream_t stream)
{
    const float* x      = (const float*)d_in[0];
    const float* conv_w = (const float*)d_in[1];
    const float* conv_b = (const float*)d_in[2];
    const float* gamma  = (const float*)d_in[3];
    const float* beta   = (const float*)d_in[4];
    float* out_xn   = (float*)d_out;
    float* out_attn = out_xn + (size_t)BATCH * CIN * NPIX;

    char* ws = (char*)d_ws;
    size_t off = 0;
    auto alloc = [&](size_t bytes) {
        size_t o = off;
        off = (off + bytes + 255) & ~(size_t)255;
        return (void*)(ws + o);
    };
    unsigned char* am   = (unsigned char*)alloc((size_t)BATCH * NPIX);
    int*   labels  = (int*)  alloc((size_t)2 * BATCH * NPIX * 4);
    int*   lists   = (int*)  alloc((size_t)16 * LISTCAP * 4);
    int*   listcnt = (int*)  alloc(16 * 4);
    int*   bgflag  = (int*)  alloc(16 * 4);
    int*   slots   = (int*)  alloc(16 * LIMIT * 4);
    unsigned char* seg = (unsigned char*)alloc((size_t)2 * BATCH * NPIX);
    float* sums    = (float*)alloc((size_t)2 * BATCH * LIMIT * CIN * 4);
    float* cnts    = (float*)alloc((size_t)2 * BATCH * LIMIT * 4);
    float* multg   = (float*)alloc((size_t)2 * BATCH * (LIMIT + 1) * 4);
    float* factor  = (float*)alloc((size_t)BATCH * NPIX * 4);
    float* chansum = (float*)alloc(CIN * 4);
    float* chansq  = (float*)alloc(CIN * 4);
    float* meaninv = (float*)alloc(2 * CIN * 4);

    (void)hipMemsetAsync(listcnt, 0, 16 * 4, stream);
    (void)hipMemsetAsync(bgflag,  0, 16 * 4, stream);
    (void)hipMemsetAsync(sums,    0, (size_t)2 * BATCH * LIMIT * CIN * 4, stream);
    (void)hipMemsetAsync(cnts,    0, (size_t)2 * BATCH * LIMIT * 4, stream);
    (void)hipMemsetAsync(chansum, 0, CIN * 4, stream);
    (void)hipMemsetAsync(chansq,  0, CIN * 4, stream);

    conv_attn_kernel<<<dim3(2, H_IMG, BATCH), 160, 0, stream>>>(x, conv_w, conv_b, out_attn, am);

    int nlab = 2 * BATCH * NPIX;
    ccl_init_kernel<<<(nlab + 255) / 256, 256, 0, stream>>>(am, labels);
    for (int r = 0; r < 20; ++r)
        ccl_prop_kernel<<<dim3(W_IMG / 32, H_IMG / 32, 16), dim3(32, 8), 0, stream>>>(labels);

    reps_kernel<<<(nlab + 255) / 256, 256, 0, stream>>>(labels, lists, listcnt, bgflag);
    topk_kernel<<<16, 256, 0, stream>>>(lists, listcnt, bgflag, slots);
    seg_kernel<<<(nlab + 255) / 256, 256, 0, stream>>>(labels, slots, seg);

    stats_kernel<<<dim3(NPIX / 256, BATCH), 256, 0, stream>>>(x, seg, sums, cnts);
    meansmult_kernel<<<BATCH, 256, 0, stream>>>(sums, cnts, multg);
    factor_kernel<<<(BATCH * NPIX + 255) / 256, 256, 0, stream>>>(seg, multg, factor);

    chanstat_kernel<<<dim3(CIN, BATCH), 256, 0, stream>>>(x, factor, chansum, chansq);
    finalize_kernel<<<1, 256, 0, stream>>>(chansum, chansq, meaninv);
    norm_kernel<<<dim3(CIN, BATCH), 256, 0, stream>>>(x, factor, meaninv, gamma, beta, out_xn);
}